// EgnnCriticNet_38448547234285
// MI455X (gfx1250) — compile-verified
//
#include <hip/hip_runtime.h>
#include <hip/hip_bf16.h>
#include <math.h>

#define N_NODES_TOT 20480
#define N_EDGES     389120
#define HID         64
#define OBS_STRIDE  10
#define BATCH       1024

typedef __attribute__((ext_vector_type(2))) float v2f;
typedef __attribute__((ext_vector_type(8))) float v8f;

// Fast activations: hardware v_rcp_f32 (~1 ulp) instead of IEEE divide chains.
__device__ __forceinline__ float fast_rcp(float v) { return __builtin_amdgcn_rcpf(v); }
__device__ __forceinline__ float sigmoidf_(float v) {
  return fast_rcp(1.0f + __expf(-v));
}
__device__ __forceinline__ float siluf(float v) { return v * sigmoidf_(v); }
__device__ __forceinline__ float tanh_fast(float v) {
  float c = fminf(fmaxf(v, -15.0f), 15.0f);
  float t = __expf(2.0f * c);
  return (t - 1.0f) * fast_rcp(t + 1.0f);
}

// LDS dot product over 64 floats using b128 loads.
__device__ __forceinline__ float dot64(const float* a, const float* w) {
  float s = 0.f;
#pragma unroll
  for (int q = 0; q < 16; ++q) {
    float4 av = ((const float4*)a)[q];
    float4 wv = ((const float4*)w)[q];
    s += av.x * wv.x + av.y * wv.y + av.z * wv.z + av.w * wv.w;
  }
  return s;
}

// ---------------------------------------------------------------------------
// WMMA fp32 16x16x4 tiled GEMM helper (wave32).
// A (LDS): 16 rows x K, row stride sa (even).  Bt (LDS): 64 rows (N) x K,
// stride sb (even), i.e. weights stored transposed for contiguous K pairs.
// Lane mapping per CDNA5 ISA 7.12.2:
//   A vgpr v : A[M = lane&15][kb + v + 2*(lane>>4)]
//   B vgpr v : B[K = kb + v + 2*(lane>>4)][N = nc*16 + (lane&15)]
//   C/D vgpr v : D[M = v + 8*(lane>>4)][N = nc*16 + (lane&15)]
// ---------------------------------------------------------------------------
__device__ __forceinline__ void wmma_gemm16(const float* A, int sa,
                                            const float* Bt, int sb, int K,
                                            v8f acc[4], int half, int l16) {
  for (int kb = 0; kb < K; kb += 4) {
    float2 av = *(const float2*)(A + l16 * sa + kb + 2 * half);
    v2f a; a.x = av.x; a.y = av.y;
#pragma unroll
    for (int nc = 0; nc < 4; ++nc) {
      float2 bv = *(const float2*)(Bt + (nc * 16 + l16) * sb + kb + 2 * half);
      v2f b; b.x = bv.x; b.y = bv.y;
      acc[nc] = __builtin_amdgcn_wmma_f32_16x16x4_f32(false, a, false, b,
                                                      (short)0, acc[nc],
                                                      false, false);
    }
  }
}

// ---------------------------------------------------------------------------
// Embedding: x = obs[:, :2];  h = obs[:, 2:10] @ emb_w + emb_b
// ---------------------------------------------------------------------------
__global__ void egnn_embed_kernel(const float* __restrict__ obs,
                                  const float* __restrict__ embw,
                                  const float* __restrict__ embb,
                                  float* __restrict__ x, float* __restrict__ h) {
  int t = blockIdx.x * blockDim.x + threadIdx.x;
  if (t >= N_NODES_TOT * HID) return;
  int i = t >> 6, j = t & 63;
  float s = embb[j];
#pragma unroll
  for (int k = 0; k < 8; ++k) s += obs[i * OBS_STRIDE + 2 + k] * embw[k * HID + j];
  h[t] = s;
  if (j < 2) x[i * 2 + j] = obs[i * OBS_STRIDE + j];
}

__global__ void zero_kernel(float* __restrict__ p, int nElem) {
  int t = blockIdx.x * blockDim.x + threadIdx.x;
  if (t < nElem) p[t] = 0.f;
}

// ---------------------------------------------------------------------------
// Edge kernel: one wave handles a tile of 16 edges; fused edge-MLP chain.
// ---------------------------------------------------------------------------
__global__ __launch_bounds__(128)
void egnn_edge_kernel(const float* __restrict__ x, const float* __restrict__ h,
                      const int* __restrict__ erow, const int* __restrict__ ecol,
                      const float* __restrict__ W1, const float* __restrict__ b1,
                      const float* __restrict__ W2, const float* __restrict__ b2,
                      const float* __restrict__ attw, const float* __restrict__ attb,
                      const float* __restrict__ cw1, const float* __restrict__ cb1,
                      const float* __restrict__ cw2,
                      float* __restrict__ agg, float* __restrict__ m_agg) {
  extern __shared__ float smem[];
  const int SA = 132, SB1 = 132, S64 = 66;
  float* sW1t  = smem;              // 64*132 (W1 transposed, K=129 in pad 132)
  float* sB1   = sW1t + 64 * SB1;   // 64
  float* sW2t  = sB1 + 64;          // 64*66
  float* sB2   = sW2t + 64 * S64;   // 64
  float* sAtt  = sB2 + 64;          // 66 ([0..63] weights, [64] bias)
  float* sCw1t = sAtt + 66;         // 64*66
  float* sCb1  = sCw1t + 64 * S64;  // 64
  float* sCw2  = sCb1 + 64;         // 64
  float* wbase = sCw2 + 64;         // per-wave scratch

  const int tid = threadIdx.x;
  for (int idx = tid; idx < 64 * 129; idx += blockDim.x) {
    int nn = idx / 129, kk = idx - nn * 129;
    sW1t[nn * SB1 + kk] = W1[kk * 64 + nn];
  }
  for (int idx = tid; idx < 64 * 64; idx += blockDim.x) {
    int nn = idx >> 6, kk = idx & 63;
    sW2t[nn * S64 + kk]  = W2[kk * 64 + nn];
    sCw1t[nn * S64 + kk] = cw1[kk * 64 + nn];
  }
  for (int idx = tid; idx < 64; idx += blockDim.x) {
    sB1[idx] = b1[idx]; sB2[idx] = b2[idx];
    sAtt[idx] = attw[idx]; sCb1[idx] = cb1[idx]; sCw2[idx] = cw2[idx];
  }
  if (tid == 0) sAtt[64] = attb[0];
  __syncthreads();

  const int wave = tid >> 5, lane = tid & 31;
  const int half = lane >> 4, l16 = lane & 15;
  const int PERW = 16 * SA + 2 * (16 * S64) + 32;   // ein + bufA + bufM + misc
  float* ein  = wbase + wave * PERW;     // 16 x 132 : [h_row | h_col | radial]
  float* bufA = ein + 16 * SA;           // 16 x 66  : m1, later p
  float* bufM = bufA + 16 * S64;         // 16 x 66  : m (gated)
  int*   rowA = (int*)(bufM + 16 * S64); // 16 edge row ids
  float* aArr = (float*)(rowA + 16);     // 16 attention gates

  const int wpb = blockDim.x >> 5;
  const int nTiles = N_EDGES / 16;       // E divisible by 16 exactly
  for (int tile = blockIdx.x * wpb + wave; tile < nTiles; tile += gridDim.x * wpb) {
    int e = tile * 16 + l16;
    int r = erow[e], c = ecol[e];
    int src = half ? c : r;
    {  // gather h rows: lane (l16,half) copies 64 floats of edge l16, part half
      const float4* hp = (const float4*)(h + (size_t)src * HID);
      float4* dp = (float4*)(ein + l16 * SA + half * 64);
#pragma unroll
      for (int q = 0; q < 16; ++q) dp[q] = hp[q];
    }
    float cdx = 0.f, cdy = 0.f;
    if (half == 0) {
      float dx = x[r * 2 + 0] - x[c * 2 + 0];
      float dy = x[r * 2 + 1] - x[c * 2 + 1];
      float radial = dx * dx + dy * dy;
      float inv = fast_rcp(sqrtf(radial) + 1e-8f);
      cdx = dx * inv; cdy = dy * inv;
      ein[l16 * SA + 128] = radial;
      rowA[l16] = r;
    }
    // ---- GEMM1: m1 = silu(e_in @ W1 + b1); K=128 via WMMA + rank-1 radial
    v8f acc[4] = {};
    wmma_gemm16(ein, SA, sW1t, SB1, 128, acc, half, l16);
#pragma unroll
    for (int nc = 0; nc < 4; ++nc) {
      float wl = sW1t[(nc * 16 + l16) * SB1 + 128];
      float bn = sB1[nc * 16 + l16];
#pragma unroll
      for (int v = 0; v < 8; ++v) {
        float rad = ein[(v + 8 * half) * SA + 128];
        bufA[(v + 8 * half) * S64 + nc * 16 + l16] =
            siluf(acc[nc][v] + rad * wl + bn);
      }
    }
    // ---- GEMM2: m = silu(m1 @ W2 + b2)
    v8f acc2[4] = {};
    wmma_gemm16(bufA, S64, sW2t, S64, 64, acc2, half, l16);
#pragma unroll
    for (int nc = 0; nc < 4; ++nc) {
      float bn = sB2[nc * 16 + l16];
#pragma unroll
      for (int v = 0; v < 8; ++v)
        bufM[(v + 8 * half) * S64 + nc * 16 + l16] = siluf(acc2[nc][v] + bn);
    }
    // ---- attention gate: m *= sigmoid(m @ att_w + att_b)
    if (half == 0) {
      float s = sAtt[64] + dot64(bufM + l16 * S64, sAtt);
      aArr[l16] = sigmoidf_(s);
    }
    {
      float sc = aArr[l16];
#pragma unroll
      for (int t = 0; t < 32; ++t) bufM[l16 * S64 + half * 32 + t] *= sc;
    }
    // ---- GEMM3: p = silu(m @ cw1 + cb1); w = tanh(p @ cw2)
    v8f accp[4] = {};
    wmma_gemm16(bufM, S64, sCw1t, S64, 64, accp, half, l16);
#pragma unroll
    for (int nc = 0; nc < 4; ++nc) {
      float bn = sCb1[nc * 16 + l16];
#pragma unroll
      for (int v = 0; v < 8; ++v)
        bufA[(v + 8 * half) * S64 + nc * 16 + l16] = siluf(accp[nc][v] + bn);
    }
    if (half == 0) {
      float wc = tanh_fast(dot64(bufA + l16 * S64, sCw2));
      unsafeAtomicAdd(&agg[r * 2 + 0], cdx * wc);
      unsafeAtomicAdd(&agg[r * 2 + 1], cdy * wc);
    }
    {  // scatter m into m_agg[row] (segment sum via native f32 atomics in L2)
      int rr = rowA[l16];
#pragma unroll
      for (int t = 0; t < 32; ++t) {
        int k = half * 32 + t;
        unsafeAtomicAdd(&m_agg[rr * HID + k], bufM[l16 * S64 + k]);
      }
    }
  }
}

// ---------------------------------------------------------------------------
// Node kernel: x += agg/19;  h += silu([h|m_agg] @ nw1 + nb1) @ nw2 + nb2
// ---------------------------------------------------------------------------
__global__ __launch_bounds__(128)
void egnn_node_kernel(float* __restrict__ x, float* __restrict__ h,
                      const float* __restrict__ agg, const float* __restrict__ m_agg,
                      const float* __restrict__ nw1, const float* __restrict__ nb1,
                      const float* __restrict__ nw2, const float* __restrict__ nb2) {
  extern __shared__ float smem[];
  const int SA = 132, S64 = 66;
  float* sW1t  = smem;             // 64*132 (K=128)
  float* sB1   = sW1t + 64 * SA;   // 64
  float* sW2t  = sB1 + 64;         // 64*66
  float* sB2   = sW2t + 64 * S64;  // 64
  float* wbase = sB2 + 64;
  const int tid = threadIdx.x;
  for (int idx = tid; idx < 64 * 128; idx += blockDim.x) {
    int nn = idx >> 7, kk = idx & 127;
    sW1t[nn * SA + kk] = nw1[kk * 64 + nn];
  }
  for (int idx = tid; idx < 64 * 64; idx += blockDim.x) {
    int nn = idx >> 6, kk = idx & 63;
    sW2t[nn * S64 + kk] = nw2[kk * 64 + nn];
  }
  for (int idx = tid; idx < 64; idx += blockDim.x) { sB1[idx] = nb1[idx]; sB2[idx] = nb2[idx]; }
  __syncthreads();

  const int wave = tid >> 5, lane = tid & 31;
  const int half = lane >> 4, l16 = lane & 15;
  const int PERW = 16 * SA + 16 * S64;
  float* hin  = wbase + wave * PERW;   // 16 x 132 : [h | m_agg]
  float* bufA = hin + 16 * SA;         // 16 x 66
  const int wpb = blockDim.x >> 5;
  const int nTiles = N_NODES_TOT / 16;
  for (int tile = blockIdx.x * wpb + wave; tile < nTiles; tile += gridDim.x * wpb) {
    int nd = tile * 16 + l16;
    {
      const float4* sp = (const float4*)((half ? m_agg : h) + (size_t)nd * HID);
      float4* dp = (float4*)(hin + l16 * SA + half * 64);
#pragma unroll
      for (int q = 0; q < 16; ++q) dp[q] = sp[q];
    }
    if (half == 0) {  // cnt == 19 for every node in the complete graph
      x[nd * 2 + 0] += agg[nd * 2 + 0] * (1.0f / 19.0f);
      x[nd * 2 + 1] += agg[nd * 2 + 1] * (1.0f / 19.0f);
    }
    v8f acc[4] = {};
    wmma_gemm16(hin, SA, sW1t, SA, 128, acc, half, l16);
#pragma unroll
    for (int nc = 0; nc < 4; ++nc) {
      float bn = sB1[nc * 16 + l16];
#pragma unroll
      for (int v = 0; v < 8; ++v)
        bufA[(v + 8 * half) * S64 + nc * 16 + l16] = siluf(acc[nc][v] + bn);
    }
    v8f acc2[4] = {};
    wmma_gemm16(bufA, S64, sW2t, S64, 64, acc2, half, l16);
#pragma unroll
    for (int nc = 0; nc < 4; ++nc) {
      float bn = sB2[nc * 16 + l16];
#pragma unroll
      for (int v = 0; v < 8; ++v) {
        int M = v + 8 * half, N = nc * 16 + l16;
        h[((size_t)tile * 16 + M) * HID + N] = hin[M * SA + N] + acc2[nc][v] + bn;
      }
    }
  }
}

// ---------------------------------------------------------------------------
// Head: z = tanh([xsq|h] @ fc1 + b); nodeval = z @ fc2 + b2
// ---------------------------------------------------------------------------
__global__ __launch_bounds__(128)
void egnn_head_kernel(const float* __restrict__ x, const float* __restrict__ h,
                      const float* __restrict__ fc1w, const float* __restrict__ fc1b,
                      const float* __restrict__ fc2w, const float* __restrict__ fc2b,
                      float* __restrict__ nodeval) {
  extern __shared__ float smem[];
  const int S64 = 66;
  float* f1t   = smem;            // 64*66 (rows 1..64 transposed)
  float* row0  = f1t + 64 * S64;  // 64 (xsq coefficients)
  float* hb1   = row0 + 64;       // 64
  float* f2    = hb1 + 64;        // 64
  float* wbase = f2 + 64;
  const int tid = threadIdx.x;
  for (int idx = tid; idx < 64 * 64; idx += blockDim.x) {
    int nn = idx >> 6, kk = idx & 63;
    f1t[nn * S64 + kk] = fc1w[(1 + kk) * 64 + nn];
  }
  for (int idx = tid; idx < 64; idx += blockDim.x) {
    row0[idx] = fc1w[idx]; hb1[idx] = fc1b[idx]; f2[idx] = fc2w[idx];
  }
  __syncthreads();
  const int wave = tid >> 5, lane = tid & 31;
  const int half = lane >> 4, l16 = lane & 15;
  const int PERW = 2 * 16 * S64 + 16;
  float* bufH = wbase + wave * PERW;
  float* bufZ = bufH + 16 * S64;
  float* xsA  = bufZ + 16 * S64;
  const int wpb = blockDim.x >> 5;
  const int nTiles = N_NODES_TOT / 16;
  for (int tile = blockIdx.x * wpb + wave; tile < nTiles; tile += gridDim.x * wpb) {
    int nd = tile * 16 + l16;
    {
      const float4* sp = (const float4*)(h + (size_t)nd * HID + half * 32);
      float4* dp = (float4*)(bufH + l16 * S64 + half * 32);
#pragma unroll
      for (int q = 0; q < 8; ++q) dp[q] = sp[q];
    }
    if (half == 0) {
      float a = x[nd * 2 + 0], b = x[nd * 2 + 1];
      xsA[l16] = a * a + b * b;
    }
    v8f acc[4] = {};
    wmma_gemm16(bufH, S64, f1t, S64, 64, acc, half, l16);
#pragma unroll
    for (int nc = 0; nc < 4; ++nc) {
      float r0 = row0[nc * 16 + l16];
      float bn = hb1[nc * 16 + l16];
#pragma unroll
      for (int v = 0; v < 8; ++v) {
        int M = v + 8 * half;
        bufZ[M * S64 + nc * 16 + l16] = tanh_fast(acc[nc][v] + xsA[M] * r0 + bn);
      }
    }
    if (half == 0) {
      nodeval[nd] = fc2b[0] + dot64(bufZ + l16 * S64, f2);
    }
  }
}

// values = per-batch mean of 20 node values; rnn passthrough
__global__ void mean_copy_kernel(const float* __restrict__ nodeval,
                                 const float* __restrict__ rnn,
                                 float* __restrict__ out) {
  int t = blockIdx.x * blockDim.x + threadIdx.x;
  if (t < BATCH) {
    float s = 0.f;
#pragma unroll
    for (int k = 0; k < 20; ++k) s += nodeval[t * 20 + k];
    out[t] = s * (1.0f / 20.0f);
  }
  if (t < BATCH * HID) out[BATCH + t] = rnn[t];
}

extern "C" void kernel_launch(void* const* d_in, const int* in_sizes, int n_in,
                              void* d_out, int out_size, void* d_ws, size_t ws_size,
                              hipStream_t stream) {
  const float* cent_obs = (const float*)d_in[0];
  const float* rnn      = (const float*)d_in[1];
  const float* emb_w    = (const float*)d_in[3];
  const float* emb_b    = (const float*)d_in[4];
  const float* edge_w1  = (const float*)d_in[5];
  const float* edge_b1  = (const float*)d_in[6];
  const float* edge_w2  = (const float*)d_in[7];
  const float* edge_b2  = (const float*)d_in[8];
  const float* att_w    = (const float*)d_in[9];
  const float* att_b    = (const float*)d_in[10];
  const float* node_w1  = (const float*)d_in[11];
  const float* node_b1  = (const float*)d_in[12];
  const float* node_w2  = (const float*)d_in[13];
  const float* node_b2  = (const float*)d_in[14];
  const float* coord_w1 = (const float*)d_in[15];
  const float* coord_b1 = (const float*)d_in[16];
  const float* coord_w2 = (const float*)d_in[17];
  const float* fc1_w    = (const float*)d_in[18];
  const float* fc1_b    = (const float*)d_in[19];
  const float* fc2_w    = (const float*)d_in[20];
  const float* fc2_b    = (const float*)d_in[21];
  const int*   erow     = (const int*)d_in[22];
  const int*   ecol     = (const int*)d_in[23];

  float* ws = (float*)d_ws;
  const int n = N_NODES_TOT;
  float* x       = ws;
  float* h       = x + (size_t)n * 2;
  float* agg     = h + (size_t)n * HID;     // agg + m_agg contiguous for zeroing
  float* m_agg   = agg + (size_t)n * 2;
  float* nodeval = m_agg + (size_t)n * HID;

  egnn_embed_kernel<<<(n * HID + 255) / 256, 256, 0, stream>>>(cent_obs, emb_w, emb_b, x, h);

  const size_t edgeSmem = (size_t)(17218 + 4 * 4256) * sizeof(float);
  const size_t nodeSmem = (size_t)(12800 + 4 * 3168) * sizeof(float);
  const size_t headSmem = (size_t)(4416 + 4 * 2128) * sizeof(float);

  for (int l = 0; l < 3; ++l) {
    zero_kernel<<<(n * 66 + 255) / 256, 256, 0, stream>>>(agg, n * 66);
    egnn_edge_kernel<<<640, 128, edgeSmem, stream>>>(
        x, h, erow, ecol,
        edge_w1 + (size_t)l * 129 * 64, edge_b1 + l * 64,
        edge_w2 + (size_t)l * 64 * 64,  edge_b2 + l * 64,
        att_w + l * 64, att_b + l,
        coord_w1 + (size_t)l * 64 * 64, coord_b1 + l * 64,
        coord_w2 + l * 64,
        agg, m_agg);
    egnn_node_kernel<<<320, 128, nodeSmem, stream>>>(
        x, h, agg, m_agg,
        node_w1 + (size_t)l * 128 * 64, node_b1 + l * 64,
        node_w2 + (size_t)l * 64 * 64,  node_b2 + l * 64);
  }
  egnn_head_kernel<<<320, 128, headSmem, stream>>>(x, h, fc1_w, fc1_b, fc2_w, fc2_b, nodeval);
  mean_copy_kernel<<<(BATCH * HID + 255) / 256, 256, 0, stream>>>(nodeval, rnn, (float*)d_out);
}